// GNN_3221225472589
// MI455X (gfx1250) — compile-verified
//
#include <hip/hip_runtime.h>
#include <hip/hip_bf16.h>

typedef __attribute__((ext_vector_type(2))) float v2f;
typedef __attribute__((ext_vector_type(8))) float v8f;

#if defined(__has_builtin)
#if __has_builtin(__builtin_amdgcn_wmma_f32_16x16x4_f32)
#define HAVE_WMMA_F32X4 1
#endif
#endif

// ---------------------------------------------------------------------------
// K0: deg[i] = 1.0f  (self-loop contributes 1 to every node's degree)
__global__ __launch_bounds__(256) void gcn_deg_init(float* __restrict__ deg, int N) {
    int i = blockIdx.x * blockDim.x + threadIdx.x;
    if (i < N) deg[i] = 1.0f;
}

// K1: deg[dst[e]] += 1
__global__ __launch_bounds__(256) void gcn_deg_accum(const long long* __restrict__ dst,
                                                     float* __restrict__ deg, long long E) {
    long long e = (long long)blockIdx.x * blockDim.x + threadIdx.x;
    if (e < E) atomicAdd(&deg[(int)dst[e]], 1.0f);
}

// K2: dinv[i] = rsqrt(deg[i])   (deg >= 1 always, no zero guard needed)
__global__ __launch_bounds__(256) void gcn_deg_rsqrt(float* __restrict__ deg, int N) {
    int i = blockIdx.x * blockDim.x + threadIdx.x;
    if (i < N) deg[i] = rsqrtf(deg[i]);
}

// ---------------------------------------------------------------------------
// K3: h1 = x @ W1   via V_WMMA_F32_16X16X4_F32.  One wave -> 16x32 output slab.
// A (16x4 f32): lane l holds A[l&15][(l>>4)*2 + {0,1}] in v2f
// B (4x16 f32): lane l holds B[(l>>4)*2 + {0,1}][l&15] in v2f (mirrors A)
// C (16x16 f32): v8f; VGPR i: lanes 0-15 -> M=i, lanes 16-31 -> M=i+8, N=lane&15
#ifdef HAVE_WMMA_F32X4
__global__ __launch_bounds__(256) void gcn_gemm1_wmma(const float* __restrict__ x,
                                                      const float* __restrict__ W1,
                                                      float* __restrict__ h1, int nTiles) {
    int wave = blockIdx.x * (blockDim.x >> 5) + (threadIdx.x >> 5);
    if (wave >= nTiles) return;                 // uniform per wave: EXEC stays all-1s
    int lane = threadIdx.x & 31;
    int row  = lane & 15;
    int half = lane >> 4;                       // 0 or 1
    int koff = half * 2;                        // K sub-offset for this half-wave

    const float* arow = x + (size_t)(wave * 16 + row) * 256;
    v8f c0 = {}; v8f c1 = {};
    for (int k = 0; k < 256; k += 4) {
        v2f a;
        a.x = arow[k + koff];
        a.y = arow[k + koff + 1];
        const float* w = W1 + (size_t)(k + koff) * 32 + row;   // col = lane&15
        v2f bt0, bt1;
        bt0.x = w[0];   bt0.y = w[32];          // cols 0..15,  K rows koff, koff+1
        bt1.x = w[16];  bt1.y = w[48];          // cols 16..31
        c0 = __builtin_amdgcn_wmma_f32_16x16x4_f32(false, a, false, bt0, (short)0, c0, false, false);
        c1 = __builtin_amdgcn_wmma_f32_16x16x4_f32(false, a, false, bt1, (short)0, c1, false, false);
    }
    // store 16x32 slab: lanes 0-15 rows 0..7, lanes 16-31 rows 8..15
    float* ob = h1 + (size_t)(wave * 16 + half * 8) * 32 + row;
#pragma unroll
    for (int i = 0; i < 8; ++i) {
        ob[(size_t)i * 32]      = c0[i];
        ob[(size_t)i * 32 + 16] = c1[i];
    }
}
#else
// VALU fallback (only if the f32x4 WMMA builtin is missing on this toolchain)
__global__ __launch_bounds__(256) void gcn_gemm1_valu(const float* __restrict__ x,
                                                      const float* __restrict__ W1,
                                                      float* __restrict__ h1, int N) {
    int t = blockIdx.x * blockDim.x + threadIdx.x;      // t = node*32 + j
    if (t >= N * 32) return;
    int i = t >> 5, j = t & 31;
    const float* xr = x + (size_t)i * 256;
    float acc = 0.f;
    for (int k = 0; k < 256; ++k) acc = fmaf(xr[k], W1[(size_t)k * 32 + j], acc);
    h1[t] = acc;
}
#endif

// ---------------------------------------------------------------------------
// K4: agg[i][j] = h1[i][j] * dinv[i]^2   (self-loop message; plain write)
__global__ __launch_bounds__(256) void gcn_selfinit(const float* __restrict__ h1,
                                                    const float* __restrict__ dinv,
                                                    float* __restrict__ agg, long long NF) {
    long long t = (long long)blockIdx.x * blockDim.x + threadIdx.x;
    if (t >= NF) return;
    int i = (int)(t >> 5);
    float s = dinv[i];
    agg[t] = h1[t] * (s * s);
}

// K5: one WAVE per edge: agg[dst] += h1[src] * norm   (32 coalesced lanes)
__global__ __launch_bounds__(256) void gcn_scatter1(const long long* __restrict__ src,
                                                    const long long* __restrict__ dst,
                                                    const float* __restrict__ dinv,
                                                    const float* __restrict__ h1,
                                                    float* __restrict__ agg, long long E) {
    long long tid = (long long)blockIdx.x * blockDim.x + threadIdx.x;
    long long e = tid >> 5;
    if (e >= E) return;
    int j = (int)(tid & 31);
    int s = (int)src[e], d = (int)dst[e];
    float nrm = dinv[s] * dinv[d];
    atomicAdd(&agg[(size_t)d * 32 + j], h1[(size_t)s * 32 + j] * nrm);
}

// K6: per node: r = relu(agg+b1); h2 = r@W2; out = b2 + h2*dinv^2 (self loop)
__global__ __launch_bounds__(256) void gcn_finish(const float* __restrict__ agg,
                                                  const float* __restrict__ b1,
                                                  const float* __restrict__ W2,
                                                  const float* __restrict__ b2,
                                                  const float* __restrict__ dinv,
                                                  float* __restrict__ h2,
                                                  float* __restrict__ out, int N) {
    int i = blockIdx.x * blockDim.x + threadIdx.x;
    if (i >= N) return;
    const float* row = agg + (size_t)i * 32;
    float a0 = 0.f, a1 = 0.f;
#pragma unroll
    for (int j = 0; j < 32; ++j) {
        float r = fmaxf(row[j] + b1[j], 0.f);
        a0 = fmaf(r, W2[j * 2 + 0], a0);
        a1 = fmaf(r, W2[j * 2 + 1], a1);
    }
    h2[(size_t)i * 2 + 0] = a0;
    h2[(size_t)i * 2 + 1] = a1;
    float s = dinv[i], sl = s * s;
    out[(size_t)i * 2 + 0] = fmaf(a0, sl, b2[0]);
    out[(size_t)i * 2 + 1] = fmaf(a1, sl, b2[1]);
}

// K7: 2 lanes per edge: out[dst] += h2[src] * norm
__global__ __launch_bounds__(256) void gcn_scatter2(const long long* __restrict__ src,
                                                    const long long* __restrict__ dst,
                                                    const float* __restrict__ dinv,
                                                    const float* __restrict__ h2,
                                                    float* __restrict__ out, long long E) {
    long long tid = (long long)blockIdx.x * blockDim.x + threadIdx.x;
    long long e = tid >> 1;
    if (e >= E) return;
    int j = (int)(tid & 1);
    int s = (int)src[e], d = (int)dst[e];
    float nrm = dinv[s] * dinv[d];
    atomicAdd(&out[(size_t)d * 2 + j], h2[(size_t)s * 2 + j] * nrm);
}

// ---------------------------------------------------------------------------
static inline int cdivll(long long a, long long b) { return (int)((a + b - 1) / b); }

extern "C" void kernel_launch(void* const* d_in, const int* in_sizes, int n_in,
                              void* d_out, int out_size, void* d_ws, size_t ws_size,
                              hipStream_t stream) {
    const float*     x  = (const float*)d_in[0];
    const long long* ei = (const long long*)d_in[1];   // int64 edge_index [2, E]
    const float*     W1 = (const float*)d_in[2];
    const float*     b1 = (const float*)d_in[3];
    const float*     W2 = (const float*)d_in[4];
    const float*     b2 = (const float*)d_in[5];

    const int       N = in_sizes[0] / 256;             // 100000
    const long long E = (long long)in_sizes[1] / 2;    // 3200000
    const long long* src = ei;
    const long long* dst = ei + E;
    float* out = (float*)d_out;

    // workspace layout (floats): dinv[N] | h1[N*32] | agg[N*32] | h2[N*2]
    float* dinv = (float*)d_ws;
    float* h1   = dinv + N;
    float* agg  = h1 + (size_t)N * 32;
    float* h2   = agg + (size_t)N * 32;

    const int T = 256;

    gcn_deg_init <<<cdivll(N, T), T, 0, stream>>>(dinv, N);
    gcn_deg_accum<<<cdivll(E, T), T, 0, stream>>>(dst, dinv, E);
    gcn_deg_rsqrt<<<cdivll(N, T), T, 0, stream>>>(dinv, N);

#ifdef HAVE_WMMA_F32X4
    {
        int nTiles = N / 16;                            // 6250 (N is 16-aligned)
        int wavesPerBlock = T / 32;                     // 8
        gcn_gemm1_wmma<<<cdivll(nTiles, wavesPerBlock), T, 0, stream>>>(x, W1, h1, nTiles);
    }
#else
    gcn_gemm1_valu<<<cdivll((long long)N * 32, T), T, 0, stream>>>(x, W1, h1, N);
#endif

    gcn_selfinit<<<cdivll((long long)N * 32, T), T, 0, stream>>>(h1, dinv, agg, (long long)N * 32);
    gcn_scatter1<<<cdivll(E * 32, T), T, 0, stream>>>(src, dst, dinv, h1, agg, E);
    gcn_finish  <<<cdivll(N, T), T, 0, stream>>>(agg, b1, W2, b2, dinv, h2, out, N);
    gcn_scatter2<<<cdivll(E * 2, T), T, 0, stream>>>(src, dst, dinv, h2, out, E);
}